// EIGNN_forward_iter_52733608461017
// MI455X (gfx1250) — compile-verified
//
#include <hip/hip_runtime.h>
#include <math.h>

typedef __attribute__((ext_vector_type(2))) float v2f;
typedef __attribute__((ext_vector_type(8))) float v8f;

#define NN 4096   // number of nodes
#define MM 64     // feature dim

// ---------------------------------------------------------------------------
// Kernel A: Gs = GAMMA * (F^T F) / (||F^T F||_F + eps)   (single workgroup)
// ---------------------------------------------------------------------------
__global__ __launch_bounds__(256) void eignn_gs_kernel(const float* __restrict__ F,
                                                       float* __restrict__ Gs) {
    __shared__ float FF[MM * MM];
    __shared__ float red[256];
    const int tid = threadIdx.x;
    float local = 0.0f;
    for (int e = tid * 16; e < tid * 16 + 16; ++e) {
        const int i = e >> 6;
        const int j = e & 63;
        float s = 0.0f;
        #pragma unroll 8
        for (int k = 0; k < MM; ++k)
            s += F[k * MM + i] * F[k * MM + j];
        FF[e] = s;
        local += s * s;
    }
    red[tid] = local;
    __syncthreads();
    for (int off = 128; off > 0; off >>= 1) {
        if (tid < off) red[tid] += red[tid + off];
        __syncthreads();
    }
    const float scale = 0.8f / (sqrtf(red[0]) + 1e-12f);
    for (int e = tid * 16; e < tid * 16 + 16; ++e)
        Gs[e] = FF[e] * scale;
}

// ---------------------------------------------------------------------------
// Kernel B: WT = (Gs @ Z)^T, stored as WT[NN][MM] (transposed so the big GEMM
// reads its A-fragments coalesced). 512 waves; each wave: one 16-col N tile,
// two 16-row M tiles. K = 64 via v_wmma_f32_16x16x4_f32.
// ---------------------------------------------------------------------------
__global__ __launch_bounds__(256) void eignn_gz_kernel(const float* __restrict__ Gs,
                                                       const float* __restrict__ Z,
                                                       float* __restrict__ WT) {
    const int w    = (blockIdx.x * blockDim.x + threadIdx.x) >> 5;  // 0..511
    const int lane = threadIdx.x & 31;
    const int lh   = lane & 15;   // position within half-wave
    const int hi   = lane >> 4;   // half-wave selector (K pair select)
    const int n0   = (w >> 1) * 16;
    const int m0   = (w & 1) * 32;

    v8f acc0 = {};
    v8f acc1 = {};
    for (int k = 0; k < MM; k += 4) {
        const int kr = k + hi * 2;
        // B fragment: lane<->N, halves hold K pairs {0,1}/{2,3}
        v2f b;
        b.x = Z[kr * NN + n0 + lh];
        b.y = Z[(kr + 1) * NN + n0 + lh];
        // A fragments from Gs (row-major 64x64): lane<->M, VGPRs hold K pair
        const float* g0 = Gs + (m0 + lh) * MM + kr;
        const float* g1 = Gs + (m0 + 16 + lh) * MM + kr;
        v2f a0, a1;
        a0.x = g0[0]; a0.y = g0[1];
        a1.x = g1[0]; a1.y = g1[1];
        acc0 = __builtin_amdgcn_wmma_f32_16x16x4_f32(false, a0, false, b,
                                                     (short)0, acc0, false, false);
        acc1 = __builtin_amdgcn_wmma_f32_16x16x4_f32(false, a1, false, b,
                                                     (short)0, acc1, false, false);
    }
    // Transposed store: element (m, n) -> WT[n * MM + m]
    const int n = n0 + lh;
    #pragma unroll
    for (int r = 0; r < 8; ++r) {
        WT[n * MM + m0 + hi * 8 + r]      = acc0[r];
        WT[n * MM + m0 + 16 + hi * 8 + r] = acc1[r];
    }
}

// ---------------------------------------------------------------------------
// Kernel C: Z = WT^T @ S + X  (the 64 x 4096 x 4096 GEMM, K = 4096).
// A-fragments from WT are coalesced (16 consecutive floats per K-row per
// half-wave); B-fragments from row-major S are coalesced along N.
// ---------------------------------------------------------------------------
__global__ __launch_bounds__(256) void eignn_ws_kernel(const float* __restrict__ WT,
                                                       const float* __restrict__ S,
                                                       const float* __restrict__ X,
                                                       float* __restrict__ Z) {
    const int w    = (blockIdx.x * blockDim.x + threadIdx.x) >> 5;  // 0..511
    const int lane = threadIdx.x & 31;
    const int lh   = lane & 15;
    const int hi   = lane >> 4;
    const int n0   = (w >> 1) * 16;
    const int m0   = (w & 1) * 32;

    v8f acc0 = {};
    v8f acc1 = {};
    const float* Scol = S + n0 + lh;   // column strip of S for this N tile
    #pragma unroll 2
    for (int k = 0; k < NN; k += 4) {
        const size_t kr = (size_t)(k + hi * 2);
        v2f b;
        b.x = Scol[kr * NN];
        b.y = Scol[(kr + 1) * NN];
        const float* w0 = WT + kr * MM;         // row kr   of WT (A column kr)
        const float* w1 = WT + (kr + 1) * MM;   // row kr+1 of WT
        v2f a0, a1;
        a0.x = w0[m0 + lh];      a0.y = w1[m0 + lh];
        a1.x = w0[m0 + 16 + lh]; a1.y = w1[m0 + 16 + lh];
        acc0 = __builtin_amdgcn_wmma_f32_16x16x4_f32(false, a0, false, b,
                                                     (short)0, acc0, false, false);
        acc1 = __builtin_amdgcn_wmma_f32_16x16x4_f32(false, a1, false, b,
                                                     (short)0, acc1, false, false);
    }
    // Epilogue: Z = acc + X, row-major coalesced stores
    const int n = n0 + lh;
    #pragma unroll
    for (int r = 0; r < 8; ++r) {
        const int ma = m0 + hi * 8 + r;
        const int mb = m0 + 16 + hi * 8 + r;
        Z[ma * NN + n] = acc0[r] + X[ma * NN + n];
        Z[mb * NN + n] = acc1[r] + X[mb * NN + n];
    }
}

// ---------------------------------------------------------------------------
// Driver: Z1 = X (since Z0 = 0), then 29 x { W = GsZ ; Z = W S + X }.
// ---------------------------------------------------------------------------
extern "C" void kernel_launch(void* const* d_in, const int* in_sizes, int n_in,
                              void* d_out, int out_size, void* d_ws, size_t ws_size,
                              hipStream_t stream) {
    (void)in_sizes; (void)n_in; (void)out_size; (void)ws_size;
    const float* X = (const float*)d_in[0];   // 64 x 4096
    const float* F = (const float*)d_in[1];   // 64 x 64
    const float* S = (const float*)d_in[2];   // 4096 x 4096
    float* Z  = (float*)d_out;                              // 64 x 4096
    float* Gs = (float*)d_ws;                               // 64 x 64
    float* WT = (float*)((char*)d_ws + MM * MM * sizeof(float)); // 4096 x 64

    eignn_gs_kernel<<<1, 256, 0, stream>>>(F, Gs);
    hipMemcpyAsync(Z, X, (size_t)MM * NN * sizeof(float),
                   hipMemcpyDeviceToDevice, stream);
    for (int it = 1; it < 30; ++it) {
        eignn_gz_kernel<<<64, 256, 0, stream>>>(Gs, Z, WT);
        eignn_ws_kernel<<<64, 256, 0, stream>>>(WT, S, X, Z);
    }
}